// LinearSelfAttention_54812372631769
// MI455X (gfx1250) — compile-verified
//
#include <hip/hip_runtime.h>
#include <hip/hip_bf16.h>

typedef __attribute__((ext_vector_type(2))) float v2f;
typedef __attribute__((ext_vector_type(8))) float v8f;

#define C_DIM 128
#define NH 4
#define HD 32
#define PTS_PER_BATCH 65536         // 256*256
#define CHUNK 128                   // points per workgroup iteration (8 waves * 16)
#define CHUNKS_PER_WG 8             // 1024 points per workgroup
#define WG_PER_BATCH (PTS_PER_BATCH / (CHUNK * CHUNKS_PER_WG)) // 64
#define NWG (4 * WG_PER_BATCH)      // 256 workgroups
#define LNR 136                     // padded LDS row stride (floats), 128-wide rows
#define KVSTR 40                    // padded kv e-stride (floats) in LDS
#define WFRAGS (32 * 8 * 32)        // ks * nt * lanes fragment-pairs per W matrix

// ---------------------------------------------------------------------------
// f32 WMMA: D(16x16) = A(16x4) * B(4x16) + C.  A/B frags are v2f per lane:
//   lane L (0-15):  M/N = L,    elems K = {0,1}
//   lane L (16-31): M/N = L-16, elems K = {2,3}
// C/D: VGPR r -> M = r (lanes 0-15) / M = r+8 (lanes 16-31), N = L%16.
// ---------------------------------------------------------------------------
__device__ __forceinline__ v8f wmma4(v2f a, v2f b, v8f c) {
  return __builtin_amdgcn_wmma_f32_16x16x4_f32(false, a, false, b, (short)0, c,
                                               false, false);
}

// 16x128 = A(16x128) * W(128x128).  A-frags are aligned b64 global loads
// (xrow pre-offset by koff); B-frags are single conflict-free ds_load_b64
// from the fragment-ordered W copy in LDS.
__device__ __forceinline__ void project_lds(const float* __restrict__ xrow,
                                            const float* WF, int lane,
                                            v8f acc[8]) {
#pragma unroll
  for (int ks = 0; ks < 32; ++ks) {
    v2f a = *(const v2f*)(xrow + 4 * ks);
#pragma unroll
    for (int nt = 0; nt < 8; ++nt) {
      v2f b = *(const v2f*)(WF + ((ks * 8 + nt) * 32 + lane) * 2);
      acc[nt] = wmma4(a, b, acc[nt]);
    }
  }
}

// Cooperative refill of one W matrix into fragment order (one-time, 64 KB).
__device__ __forceinline__ void stage_wfrag(const float* __restrict__ W,
                                            float* WF, int tid) {
  for (int i = tid; i < WFRAGS; i += 256) {
    const int l   = i & 31;
    const int nt  = (i >> 5) & 7;
    const int ks  = i >> 8;
    const int kk  = 4 * ks + ((l >> 4) << 1);
    const int col = nt * 16 + (l & 15);
    WF[2 * i]     = W[kk * C_DIM + col];
    WF[2 * i + 1] = W[(kk + 1) * C_DIM + col];
  }
}

// Joint LayerNorm over the 128 columns of each point, then store to LDS.
// Point p = r + 8*half lives in VGPR r of the half-wave: reduce 8 partials
// per lane, then butterfly over the 16 lanes of each half.
__device__ __forceinline__ void ln_store(const v8f acc[8], const float* g,
                                         const float* bt, float* dst,
                                         int lm, int half) {
  float s[8], ss[8];
#pragma unroll
  for (int r = 0; r < 8; ++r) {
    s[r] = 0.f; ss[r] = 0.f;
#pragma unroll
    for (int nt = 0; nt < 8; ++nt) {
      float v = acc[nt][r];
      s[r] += v; ss[r] += v * v;
    }
  }
#pragma unroll
  for (int m = 1; m <= 8; m <<= 1) {
#pragma unroll
    for (int r = 0; r < 8; ++r) {
      s[r]  += __shfl_xor(s[r],  m, 32);
      ss[r] += __shfl_xor(ss[r], m, 32);
    }
  }
#pragma unroll
  for (int r = 0; r < 8; ++r) {
    const float mu  = s[r] * (1.0f / 128.0f);
    const float var = ss[r] * (1.0f / 128.0f) - mu * mu;
    const float rs  = rsqrtf(var + 1e-5f);
    const int row = r + 8 * half;
#pragma unroll
    for (int nt = 0; nt < 8; ++nt)
      dst[row * LNR + nt * 16 + lm] = (acc[nt][r] - mu) * rs * g[nt] + bt[nt];
  }
}

// ---------------------------------------------------------------------------
// Pass 1: k,v projections + LayerNorm + kv = k^T v accumulation (all WMMA).
// ---------------------------------------------------------------------------
__global__ void __launch_bounds__(256)
lsa_kv_kernel(const float* __restrict__ X, const float* __restrict__ Wk,
              const float* __restrict__ Wv, const float* __restrict__ gamma,
              const float* __restrict__ beta, float* __restrict__ kvg) {
  extern __shared__ float smem[];
  float* WkF = smem;                       // [WFRAGS][2]
  float* WvF = WkF + 2 * WFRAGS;           // [WFRAGS][2]
  float* kS  = WvF + 2 * WFRAGS;           // [CHUNK][LNR]
  float* vS  = kS + CHUNK * LNR;           // [CHUNK][LNR]

  const int tid  = threadIdx.x;
  const int wave = tid >> 5;
  const int lane = tid & 31;
  const int lm   = lane & 15;
  const int half = lane >> 4;
  const int koff = half * 2;

  const int b     = blockIdx.x / WG_PER_BATCH;
  const int pbase = (blockIdx.x % WG_PER_BATCH) * (CHUNK * CHUNKS_PER_WG);

  stage_wfrag(Wk, WkF, tid);
  stage_wfrag(Wv, WvF, tid);

  float g[8], bt[8];
#pragma unroll
  for (int nt = 0; nt < 8; ++nt) {
    g[nt]  = gamma[nt * 16 + lm];
    bt[nt] = beta[nt * 16 + lm];
  }
  __syncthreads();

  // Each wave owns two fixed (head, d-tile, e-tile) kv output tiles.
  const int c0 = wave * 2;
  v8f kvacc[2];
  kvacc[0] = (v8f)(0.0f);
  kvacc[1] = (v8f)(0.0f);

  for (int ch = 0; ch < CHUNKS_PER_WG; ++ch) {
    const int p0   = pbase + ch * CHUNK;
    const int prow = p0 + wave * 16 + lm;
    const float* xrow =
        X + ((size_t)b * PTS_PER_BATCH + prow) * C_DIM + koff;

    v8f acc[8];
#pragma unroll
    for (int nt = 0; nt < 8; ++nt) acc[nt] = (v8f)(0.0f);
    project_lds(xrow, WkF, lane, acc);
    ln_store(acc, g, bt, kS + wave * 16 * LNR, lm, half);

#pragma unroll
    for (int nt = 0; nt < 8; ++nt) acc[nt] = (v8f)(0.0f);
    project_lds(xrow, WvF, lane, acc);
    ln_store(acc, g, bt, vS + wave * 16 * LNR, lm, half);

    __syncthreads();

    // kv += k^T v over this chunk's 128 points (K dim = points).
#pragma unroll
    for (int j = 0; j < 2; ++j) {
      const int c = c0 + j;
      const int h = c >> 2, mt = (c >> 1) & 1, n2 = c & 1;
      v8f o = kvacc[j];
#pragma unroll
      for (int ks = 0; ks < 32; ++ks) {
        const int kk = 4 * ks + koff;
        v2f a, bf;
        a.x  = kS[kk * LNR + h * HD + mt * 16 + lm];
        a.y  = kS[(kk + 1) * LNR + h * HD + mt * 16 + lm];
        bf.x = vS[kk * LNR + h * HD + n2 * 16 + lm];
        bf.y = vS[(kk + 1) * LNR + h * HD + n2 * 16 + lm];
        o = wmma4(a, bf, o);
      }
      kvacc[j] = o;
    }
    __syncthreads();
  }

  // One atomic flush per wave for its two 16x16 kv tiles.
#pragma unroll
  for (int j = 0; j < 2; ++j) {
    const int c = c0 + j;
    const int h = c >> 2, mt = (c >> 1) & 1, n2 = c & 1;
#pragma unroll
    for (int r = 0; r < 8; ++r) {
      const int d = mt * 16 + r + 8 * half;
      const int e = n2 * 16 + lm;
      atomicAdd(&kvg[(((size_t)b * NH + h) * HD + d) * HD + e], kvacc[j][r]);
    }
  }
}

// ---------------------------------------------------------------------------
// Pass 2: q projection (X is L2-resident) + sa = q @ kv, scaled, interleaved.
// ---------------------------------------------------------------------------
__global__ void __launch_bounds__(256)
lsa_out_kernel(const float* __restrict__ X, const float* __restrict__ Wq,
               const float* __restrict__ kvg, float* __restrict__ out) {
  extern __shared__ float smem[];
  float* WqF = smem;                        // [WFRAGS][2]
  float* kvS = WqF + 2 * WFRAGS;            // [NH][HD][KVSTR]
  float* qS  = kvS + NH * HD * KVSTR;       // [8 waves][16][LNR]

  const int tid  = threadIdx.x;
  const int wave = tid >> 5;
  const int lane = tid & 31;
  const int lm   = lane & 15;
  const int half = lane >> 4;
  const int koff = half * 2;

  const int b     = blockIdx.x / WG_PER_BATCH;
  const int pbase = (blockIdx.x % WG_PER_BATCH) * (CHUNK * CHUNKS_PER_WG);

  stage_wfrag(Wq, WqF, tid);

  // Stage this batch's kv (16 KB) into padded LDS.
  for (int i = tid; i < NH * HD * HD; i += 256) {
    const int h = i >> 10, d = (i >> 5) & 31, e = i & 31;
    kvS[(h * HD + d) * KVSTR + e] = kvg[(size_t)b * NH * HD * HD + i];
  }
  __syncthreads();

  float* qw = qS + wave * 16 * LNR;
  const float inv_n = 1.0f / (float)PTS_PER_BATCH;

  for (int ch = 0; ch < CHUNKS_PER_WG; ++ch) {
    const int p0   = pbase + ch * CHUNK;
    const int prow = p0 + wave * 16 + lm;
    const float* xrow =
        X + ((size_t)b * PTS_PER_BATCH + prow) * C_DIM + koff;

    v8f acc[8];
#pragma unroll
    for (int nt = 0; nt < 8; ++nt) acc[nt] = (v8f)(0.0f);
    project_lds(xrow, WqF, lane, acc);

    // Round-trip q through wave-private LDS to re-fragment for the 2nd GEMM.
#pragma unroll
    for (int r = 0; r < 8; ++r) {
      const int row = r + 8 * half;
#pragma unroll
      for (int nt = 0; nt < 8; ++nt)
        qw[row * LNR + nt * 16 + lm] = acc[nt][r];
    }
    __syncthreads();

    // out[p][e*4+h] = (1/N) * sum_d q[p][h,d] * kv[h][d][e]
#pragma unroll
    for (int h = 0; h < NH; ++h) {
#pragma unroll
      for (int n2 = 0; n2 < 2; ++n2) {
        v8f o = (v8f)(0.0f);
#pragma unroll
        for (int ks = 0; ks < 8; ++ks) {
          const int kk = 4 * ks + koff;
          v2f a = *(const v2f*)(qw + lm * LNR + h * HD + kk);
          v2f bf;
          bf.x = kvS[(h * HD + kk) * KVSTR + n2 * 16 + lm];
          bf.y = kvS[(h * HD + kk + 1) * KVSTR + n2 * 16 + lm];
          o = wmma4(a, bf, o);
        }
#pragma unroll
        for (int r = 0; r < 8; ++r) {
          const int pt = p0 + wave * 16 + r + 8 * half;
          const int e  = n2 * 16 + lm;
          out[((size_t)b * PTS_PER_BATCH + pt) * C_DIM + e * NH + h] =
              o[r] * inv_n;
        }
      }
    }
    __syncthreads();
  }
}

__global__ void lsa_zero_kernel(float* __restrict__ p, int n) {
  const int i = blockIdx.x * blockDim.x + threadIdx.x;
  if (i < n) p[i] = 0.0f;
}

extern "C" void kernel_launch(void* const* d_in, const int* in_sizes, int n_in,
                              void* d_out, int out_size, void* d_ws,
                              size_t ws_size, hipStream_t stream) {
  (void)in_sizes; (void)n_in; (void)out_size; (void)ws_size;
  const float* X     = (const float*)d_in[0];
  const float* Wq    = (const float*)d_in[1];
  const float* Wk    = (const float*)d_in[2];
  const float* Wv    = (const float*)d_in[3];
  const float* gamma = (const float*)d_in[4];
  const float* beta  = (const float*)d_in[5];
  float* out = (float*)d_out;
  float* kvg = (float*)d_ws;   // 4*4*32*32 fp32 accumulator = 64 KiB

  const int kvN = 4 * NH * HD * HD;
  const int lds1 =
      (4 * WFRAGS + 2 * CHUNK * LNR) * (int)sizeof(float);           // ~264 KiB
  const int lds2 =
      (2 * WFRAGS + NH * HD * KVSTR + 8 * 16 * LNR) * (int)sizeof(float); // ~152 KiB
  (void)hipFuncSetAttribute((const void*)lsa_kv_kernel,
                            hipFuncAttributeMaxDynamicSharedMemorySize, lds1);
  (void)hipFuncSetAttribute((const void*)lsa_out_kernel,
                            hipFuncAttributeMaxDynamicSharedMemorySize, lds2);

  lsa_zero_kernel<<<(kvN + 255) / 256, 256, 0, stream>>>(kvg, kvN);
  lsa_kv_kernel<<<NWG, 256, lds1, stream>>>(X, Wk, Wv, gamma, beta, kvg);
  lsa_out_kernel<<<NWG, 256, lds2, stream>>>(X, Wq, kvg, out);
}